// H2I_33758442947037
// MI455X (gfx1250) — compile-verified
//
#include <hip/hip_runtime.h>
#include <hip/hip_bf16.h>
#include <stdint.h>

#define IM_SIZE 512
#define RADIUS  16
#define PAD_VAL (-1000.0f)

// One block (128 threads, 4 wave32 waves) per image row.
//  - Stage the 512-float row into LDS with 128 async global->LDS B128 copies
//    (gfx1250 ASYNCcnt path), pad 16 floats of -1000 from registers.
//  - Each thread computes 4 consecutive outputs: their 17-tap windows overlap,
//    so 5 x ds_load_b128 (20 floats) feed all 4 outputs from registers.
//  - One global_store_b128 per thread (fully coalesced 16B/lane).
__global__ __launch_bounds__(128) void horizon_shadow_kernel(
    const float* __restrict__ in, float* __restrict__ out)
{
    __shared__ __align__(16) float smem[IM_SIZE + RADIUS];   // 528 floats

    const int row = blockIdx.x;                  // 0 .. 8191  (B*C*H)
    const int t   = threadIdx.x;                 // 0 .. 127
    const size_t row_base = (size_t)row * IM_SIZE;

    // ---- async stage: 128 threads x 16B = 2048B = full row ----------------
    {
        // LDS byte address: flat shared address truncates to the LDS offset.
        uint32_t lds_addr = (uint32_t)(uintptr_t)(&smem[0]) + (uint32_t)(t * 16);
        const float* gaddr = in + row_base + (size_t)(t * 4);
        asm volatile("global_load_async_to_lds_b128 %0, %1, off"
                     :: "v"(lds_addr), "v"(gaddr)
                     : "memory");
    }
    // Pad filled from registers (must not read past the row in global memory:
    // rows are contiguous, a global pad read would alias the next row).
    if (t < RADIUS) {
        smem[IM_SIZE + t] = PAD_VAL;
    }

    // Wait for this wave's async copies to land in LDS, then make all waves'
    // data visible workgroup-wide.
    asm volatile("s_wait_asynccnt 0" ::: "memory");
    __syncthreads();

    // ---- compute: 4 consecutive outputs per thread ------------------------
    const int w0 = t * 4;                        // 16B-aligned window base

    // 20 consecutive floats cover x[w0 .. w0+19]: 5 x ds_load_b128.
    float xa[20];
    {
        const float4* s4 = reinterpret_cast<const float4*>(&smem[w0]);
#pragma unroll
        for (int i = 0; i < 5; ++i) {
            float4 q = s4[i];
            xa[i * 4 + 0] = q.x;
            xa[i * 4 + 1] = q.y;
            xa[i * 4 + 2] = q.z;
            xa[i * 4 + 3] = q.w;
        }
    }

    float res[4];
#pragma unroll
    for (int i = 0; i < 4; ++i) {
        // m = max_{r=1..16} ( x[w+r] - r/10 ); fmaxf chain -> v_max3_f32.
        float m = xa[i + 1] - 0.1f;
#pragma unroll
        for (int r = 2; r <= RADIUS; ++r) {
            m = fmaxf(m, xa[i + r] - 0.1f * (float)r);
        }
        // 1 - clip(m - x, 0, 1); clamp lowers to v_med3_f32.
        float d = m - xa[i];
        d = fminf(fmaxf(d, 0.0f), 1.0f);
        res[i] = 1.0f - d;
    }

    float4 o;
    o.x = res[0]; o.y = res[1]; o.z = res[2]; o.w = res[3];
    *reinterpret_cast<float4*>(out + row_base + w0) = o;   // global_store_b128
}

extern "C" void kernel_launch(void* const* d_in, const int* in_sizes, int n_in,
                              void* d_out, int out_size, void* d_ws, size_t ws_size,
                              hipStream_t stream)
{
    (void)in_sizes; (void)n_in; (void)d_ws; (void)ws_size; (void)out_size;

    const float* hf  = (const float*)d_in[0];   // [16,1,512,512] fp32
    float*       img = (float*)d_out;           // [16,1,512,512] fp32

    const int n_rows = 16 * 1 * IM_SIZE;        // 8192 rows
    horizon_shadow_kernel<<<dim3(n_rows), dim3(128), 0, stream>>>(hf, img);
}